// AttentionModule_27565100105751
// MI455X (gfx1250) — compile-verified
//
#include <hip/hip_runtime.h>
#include <math.h>

// ---------------------------------------------------------------------------
// Problem constants (reference: B=4, S=2048, D=1024, R=128, D_ATT=64, D_P=64)
// ---------------------------------------------------------------------------
#define BB   4
#define SS   2048
#define DD   1024
#define RR   128
#define DATT 64
#define DP   64

typedef __attribute__((ext_vector_type(2))) float v2f;
typedef __attribute__((ext_vector_type(8))) float v8f;

// ---------------------------------------------------------------------------
// Workspace layout (float element offsets).  Total ~44.57M floats (~179 MB).
// ---------------------------------------------------------------------------
constexpr size_t OFF_QR   = 0;                                   // [B,S,R]
constexpr size_t OFF_KR   = OFF_QR  + (size_t)BB * SS * RR;      // [B,S,R]
constexpr size_t OFF_SC   = OFF_KR  + (size_t)BB * SS * RR;      // [B,S,S] scores -> probs
constexpr size_t OFF_Q    = OFF_SC  + (size_t)BB * SS * SS;      // [B,S,D] q -> ctx
constexpr size_t OFF_K    = OFF_Q   + (size_t)BB * SS * DD;      // [B,S,D]
constexpr size_t OFF_V    = OFF_K   + (size_t)BB * SS * DD;      // [B,S,D]
constexpr size_t OFF_SUM  = OFF_V   + (size_t)BB * SS * DD;      // [B,D]
constexpr size_t OFF_ZPM  = OFF_SUM + (size_t)BB * DD;           // [B,DP]
constexpr size_t OFF_MISC = OFF_ZPM + (size_t)BB * DP;           // [0]=k_eff(int) [1]=diff(u32)
constexpr size_t OFF_MASK = OFF_MISC + 16;                       // B*S*(S/32) u32 bitmask

// ---------------------------------------------------------------------------
// Batched f32 WMMA GEMM:  C[b] = alpha * A[b] @ (TRANSB ? B[b]^T : B[b])
// V_WMMA_F32_16X16X4_F32.  One wave -> 32x64 C tile (2x4 fragments, 8 accs),
// block of 4 waves -> 128x64 C tile.  TRANSB is a template param so the K
// loop is branch-free.  Requires M % 128 == 0, N % 64 == 0, K % 4 == 0
// (true for every GEMM in this pipeline).
// Lane layouts per CDNA5 ISA 7.12.2:
//   A 16x4 : lane.lo = M row, VGPR pair = K {2*hi, 2*hi+1}
//   B 4x16 : lanes = N col,  VGPR pair = K rows {2*hi, 2*hi+1}
//   C 16x16: VGPR r -> row r + 8*hi, col = lane.lo
// ---------------------------------------------------------------------------
template <int TRANSB>
__global__ __launch_bounds__(128) void wmma_gemm_f32(
    const float* __restrict__ A, const float* __restrict__ B,
    float* __restrict__ C,
    int M, int N, int K,
    long long sA, long long sB, long long sC,
    int lda, int ldb, int ldc, float alpha)
{
    const int lane = threadIdx.x & 31;
    const int wave = threadIdx.x >> 5;
    const int lo   = lane & 15;
    const int hi   = lane >> 4;

    const int n0 = blockIdx.x * 64;
    const int m0 = blockIdx.y * 128 + wave * 32;

    const float* __restrict__ Ab = A + (size_t)blockIdx.z * sA;
    const float* __restrict__ Bb = B + (size_t)blockIdx.z * sB;
    float* __restrict__       Cb = C + (size_t)blockIdx.z * sC;

    const float* __restrict__ arow0 = Ab + (size_t)(m0 + lo) * lda + 2 * hi;
    const float* __restrict__ arow1 = arow0 + (size_t)16 * lda;

    // Hoisted B base pointers (k-invariant part).
    const float* __restrict__ bbase[4];
#pragma unroll
    for (int i = 0; i < 4; ++i) {
        if (TRANSB)
            bbase[i] = Bb + (size_t)(n0 + 16 * i + lo) * ldb + 2 * hi;   // row of B^T
        else
            bbase[i] = Bb + (size_t)(2 * hi) * ldb + n0 + 16 * i + lo;   // col of B
    }

    v8f acc[2][4] = {};

    for (int k0 = 0; k0 < K; k0 += 4) {
        v2f a0, a1, bf[4];
        a0.x = arow0[k0]; a0.y = arow0[k0 + 1];
        a1.x = arow1[k0]; a1.y = arow1[k0 + 1];
        if (TRANSB) {
#pragma unroll
            for (int i = 0; i < 4; ++i) {
                bf[i].x = bbase[i][k0];
                bf[i].y = bbase[i][k0 + 1];
            }
        } else {
            const size_t koff = (size_t)k0 * ldb;
#pragma unroll
            for (int i = 0; i < 4; ++i) {
                bf[i].x = bbase[i][koff];
                bf[i].y = bbase[i][koff + (size_t)ldb];
            }
        }
#pragma unroll
        for (int i = 0; i < 4; ++i) {
            acc[0][i] = __builtin_amdgcn_wmma_f32_16x16x4_f32(
                false, a0, false, bf[i], (short)0, acc[0][i], false, false);
            acc[1][i] = __builtin_amdgcn_wmma_f32_16x16x4_f32(
                false, a1, false, bf[i], (short)0, acc[1][i], false, false);
        }
    }

#pragma unroll
    for (int mi = 0; mi < 2; ++mi) {
#pragma unroll
        for (int i = 0; i < 4; ++i) {
            float* crow = Cb + (size_t)(m0 + 16 * mi + 8 * hi) * ldc + n0 + 16 * i + lo;
#pragma unroll
            for (int r = 0; r < 8; ++r)
                crow[(size_t)r * ldc] = alpha * acc[mi][i][r];
        }
    }
}

// ---------------------------------------------------------------------------
// summary = mean_S(mamba_out)  [B,D];  zpm = mean_S(z_purp)  [B,DP]
// ---------------------------------------------------------------------------
__global__ void reduce_means_kernel(const float* __restrict__ mamba,
                                    const float* __restrict__ zp,
                                    float* __restrict__ ws)
{
    const int idx = blockIdx.x * blockDim.x + threadIdx.x;
    const float invS = 1.0f / (float)SS;
    if (idx < BB * DD) {
        const int b = idx / DD, d = idx % DD;
        const float* p = mamba + (size_t)b * SS * DD + d;
        float s = 0.f;
        for (int t = 0; t < SS; ++t) s += p[(size_t)t * DD];
        ws[OFF_SUM + idx] = s * invS;
    } else if (idx < BB * DD + BB * DP) {
        const int j = idx - BB * DD;
        const int b = j / DP, d = j % DP;
        const float* p = zp + (size_t)b * SS * DP + d;
        float s = 0.f;
        for (int t = 0; t < SS; ++t) s += p[(size_t)t * DP];
        ws[OFF_ZPM + j] = s * invS;
    }
}

// ---------------------------------------------------------------------------
// z_att = (1-w) + w*sigmoid(summary@Wgain + bgain + purpose + sal@Wsal + bsal)
// Also: k_eff = max(1, int(S*0.125*mean(z_cap/(z_cap+1e-6)))) -> ws, and
// zero the switch-loss diff counter (graph-replay safe init).
// ---------------------------------------------------------------------------
__global__ void zatt_kernel(const float* __restrict__ sal,
                            const float* __restrict__ z_cap,
                            const float* __restrict__ warmup,
                            const float* __restrict__ Wgain,
                            const float* __restrict__ bgain,
                            const float* __restrict__ Wsal,
                            const float* __restrict__ bsal,
                            const float* __restrict__ Wpurp,
                            const float* __restrict__ bpurp,
                            float* __restrict__ ws,
                            float* __restrict__ out_zatt)
{
    const int tid = threadIdx.x;            // 256 = B*DATT
    const int b = tid / DATT, j = tid % DATT;

    float acc = bgain[j] + bpurp[j] + bsal[j] + sal[b] * Wsal[j];
    const float* sm = ws + OFF_SUM + (size_t)b * DD;
    for (int d = 0; d < DD; ++d) acc += sm[d] * Wgain[d * DATT + j];
    const float* zm = ws + OFF_ZPM + (size_t)b * DP;
    for (int p = 0; p < DP; ++p) acc += zm[p] * Wpurp[p * DATT + j];

    const float learned = 1.0f / (1.0f + __expf(-acc));
    const float w = warmup[0];
    out_zatt[tid] = (1.0f - w) + w * learned;

    if (tid == 0) {
        float m = 0.f;
        for (int bb = 0; bb < BB; ++bb) m += z_cap[bb] / (z_cap[bb] + 1e-6f);
        m *= (1.0f / (float)BB);
        int keff = (int)((float)SS * 0.125f * m);   // truncation, like int()
        if (keff < 1) keff = 1;
        ((int*)(ws + OFF_MISC))[0] = keff;
        ((unsigned*)(ws + OFF_MISC))[1] = 0u;
    }
}

// ---------------------------------------------------------------------------
// Per-row top-k -> bitmask + switch-loss diff count.
// One block per (b,s) row.  Radix-select (8 x 4-bit passes) on monotone
// uint-mapped causal scores (-inf beyond the diagonal matches jax top_k's
// padding semantics); ties broken by ascending index via block scan.
// ---------------------------------------------------------------------------
__global__ void topk_mask_kernel(float* __restrict__ ws,
                                 const unsigned char* __restrict__ prev)
{
    __shared__ unsigned key[SS];
    __shared__ unsigned cnt[16];
    __shared__ unsigned sc[256];
    __shared__ unsigned mwords[SS / 32];
    __shared__ unsigned gG, gdiff;

    const int row = blockIdx.x;             // row = b*S + s
    const int s   = row % SS;
    const int tid = threadIdx.x;
    const int CH  = SS / 256;               // 8 contiguous elements / thread
    const int t0  = tid * CH;

    const float* srow = ws + OFF_SC + (size_t)row * SS;

    for (int j = 0; j < CH; ++j) {
        const int t = t0 + j;
        const float v = (t <= s) ? srow[t] : -__builtin_inff();
        unsigned u = __float_as_uint(v);
        u = (u & 0x80000000u) ? ~u : (u | 0x80000000u);   // order-preserving
        key[t] = u;
    }
    if (tid < SS / 32) mwords[tid] = 0u;
    if (tid == 0) { gG = 0u; gdiff = 0u; }
    __syncthreads();

    const int keff = ((const int*)(ws + OFF_MISC))[0];

    // radix select the keff-th largest key
    unsigned prefix = 0u, pmask = 0u;
    int kk = keff;
    for (int pos = 28; pos >= 0; pos -= 4) {
        if (tid < 16) cnt[tid] = 0u;
        __syncthreads();
        for (int j = 0; j < CH; ++j) {
            const unsigned u = key[t0 + j];
            if ((u & pmask) == prefix) atomicAdd(&cnt[(u >> pos) & 15], 1u);
        }
        __syncthreads();
        unsigned c = 0; int bsel = 0;
        for (int q = 15; q >= 0; --q) {
            const unsigned nc = c + cnt[q];
            if ((int)nc >= kk) { bsel = q; break; }
            c = nc;
        }
        kk -= (int)c;
        prefix |= ((unsigned)bsel) << pos;
        pmask  |= 0xFu << pos;
        __syncthreads();
    }
    const unsigned T = prefix;

    // strictly-greater count and per-thread tie count
    unsigned gcount = 0, tcount = 0;
    for (int j = 0; j < CH; ++j) {
        const unsigned u = key[t0 + j];
        if (u > T) gcount++;
        else if (u == T) tcount++;
    }
    atomicAdd(&gG, gcount);

    // exclusive scan of tie counts in t-order
    sc[tid] = tcount;
    __syncthreads();
    for (int off = 1; off < 256; off <<= 1) {
        const unsigned v = (tid >= off) ? sc[tid - off] : 0u;
        __syncthreads();
        sc[tid] += v;
        __syncthreads();
    }
    const unsigned excl = sc[tid] - tcount;
    __syncthreads();

    const int ties_needed = keff - (int)gG;

    unsigned mybits = 0u;
    unsigned rank = excl;
    for (int j = 0; j < CH; ++j) {
        const unsigned u = key[t0 + j];
        bool m = false;
        if (u > T) m = true;
        else if (u == T) { if ((int)rank < ties_needed) m = true; rank++; }
        if (m) mybits |= 1u << j;
    }
    atomicOr(&mwords[t0 >> 5], mybits << (t0 & 31));

    // switch-loss partial: mask != prev_mask (booleans => squared diff == xor)
    const unsigned char* prow = prev + (size_t)row * SS;
    unsigned d = 0;
    for (int j = 0; j < CH; ++j) {
        const unsigned m  = (mybits >> j) & 1u;
        const unsigned pv = prow[t0 + j] ? 1u : 0u;
        d += (m ^ pv);
    }
    atomicAdd(&gdiff, d);
    __syncthreads();

    unsigned* gmask = ((unsigned*)(ws + OFF_MASK)) + (size_t)row * (SS / 32);
    if (tid < SS / 32) gmask[tid] = mwords[tid];
    if (tid == 0) atomicAdd(((unsigned*)(ws + OFF_MISC)) + 1, gdiff);
}

// ---------------------------------------------------------------------------
// In-place masked softmax over each (b,s) row of attn scores.
// allowed = (t <= s) && maskbit.
// ---------------------------------------------------------------------------
__global__ void softmax_mask_kernel(float* __restrict__ ws)
{
    __shared__ float red[256];
    const int row = blockIdx.x;
    const int s   = row % SS;
    const int tid = threadIdx.x;
    const int CH  = SS / 256;
    const int t0  = tid * CH;

    float* srow = ws + OFF_SC + (size_t)row * SS;
    const unsigned* mrow = ((const unsigned*)(ws + OFF_MASK)) + (size_t)row * (SS / 32);
    const unsigned mbits = (mrow[t0 >> 5] >> (t0 & 31)) & 0xFFu;

    float vals[CH];
    float mx = -__builtin_inff();
    for (int j = 0; j < CH; ++j) {
        const int t = t0 + j;
        const bool allowed = (t <= s) && ((mbits >> j) & 1u);
        const float v = allowed ? srow[t] : -__builtin_inff();
        vals[j] = v;
        mx = fmaxf(mx, v);
    }
    red[tid] = mx; __syncthreads();
    for (int off = 128; off > 0; off >>= 1) {
        if (tid < off) red[tid] = fmaxf(red[tid], red[tid + off]);
        __syncthreads();
    }
    mx = red[0]; __syncthreads();

    float sum = 0.f;
    for (int j = 0; j < CH; ++j) {
        const float e = (vals[j] == -__builtin_inff()) ? 0.f : __expf(vals[j] - mx);
        vals[j] = e;
        sum += e;
    }
    red[tid] = sum; __syncthreads();
    for (int off = 128; off > 0; off >>= 1) {
        if (tid < off) red[tid] += red[tid + off];
        __syncthreads();
    }
    const float inv = 1.0f / red[0];
    for (int j = 0; j < CH; ++j) srow[t0 + j] = vals[j] * inv;
}

// ---------------------------------------------------------------------------
// switch_loss = diff_count / (B*S*S)
// ---------------------------------------------------------------------------
__global__ void finalize_loss_kernel(const float* __restrict__ ws,
                                     float* __restrict__ out_loss)
{
    const unsigned d = ((const unsigned*)(ws + OFF_MISC))[1];
    out_loss[0] = (float)d / (float)((size_t)BB * SS * SS);
}

// ---------------------------------------------------------------------------
// Launch.  Inputs in setup_inputs() order; output = [out | z_att | loss].
// Requires ws_size >= (OFF_MASK + B*S*S/32) * 4 bytes (~179 MB).
// ---------------------------------------------------------------------------
extern "C" void kernel_launch(void* const* d_in, const int* in_sizes, int n_in,
                              void* d_out, int out_size, void* d_ws, size_t ws_size,
                              hipStream_t stream)
{
    const float* x      = (const float*)d_in[0];
    const float* mamba  = (const float*)d_in[1];
    const float* sal    = (const float*)d_in[2];
    const float* zp     = (const float*)d_in[3];
    const float* zcap   = (const float*)d_in[4];
    const float* warm   = (const float*)d_in[5];
    const unsigned char* prev = (const unsigned char*)d_in[6];
    // d_in[7] = step (unused: SPARSE_FROM == 0 -> sparse branch always taken)
    const float* Wq_rel = (const float*)d_in[8];
    const float* Wk_rel = (const float*)d_in[9];
    const float* Wgain  = (const float*)d_in[10];
    const float* bgain  = (const float*)d_in[11];
    const float* Wsal   = (const float*)d_in[12];
    const float* bsal   = (const float*)d_in[13];
    const float* Wpurp  = (const float*)d_in[14];
    const float* bpurp  = (const float*)d_in[15];
    const float* Wq     = (const float*)d_in[16];
    const float* Wk     = (const float*)d_in[17];
    const float* Wv     = (const float*)d_in[18];
    const float* Wout   = (const float*)d_in[19];

    float* out = (float*)d_out;
    float* ws  = (float*)d_ws;

    float* out_zatt = out + (size_t)BB * SS * DD;
    float* out_loss = out_zatt + (size_t)BB * DATT;

    const dim3 blk(128);   // 4 waves of 32; block tile = 128 x 64
    const float inv_sqrtR = 1.0f / sqrtf((float)RR);
    const float inv_sqrtD = 1.0f / sqrtf((float)DD);

    // 1) means + gate (also computes k_eff on-device, zeroes diff counter)
    reduce_means_kernel<<<(BB * (DD + DP) + 255) / 256, 256, 0, stream>>>(mamba, zp, ws);
    zatt_kernel<<<1, 256, 0, stream>>>(sal, zcap, warm, Wgain, bgain, Wsal, bsal,
                                       Wpurp, bpurp, ws, out_zatt);

    // 2) qr / kr = mamba_out @ W[1024x128]
    wmma_gemm_f32<0><<<dim3(RR / 64, (BB * SS) / 128, 1), blk, 0, stream>>>(
        mamba, Wq_rel, ws + OFF_QR, BB * SS, RR, DD, 0, 0, 0, DD, RR, RR, 1.0f);
    wmma_gemm_f32<0><<<dim3(RR / 64, (BB * SS) / 128, 1), blk, 0, stream>>>(
        mamba, Wk_rel, ws + OFF_KR, BB * SS, RR, DD, 0, 0, 0, DD, RR, RR, 1.0f);

    // 3) scores = qr @ kr^T / sqrt(R)   [B,S,S]
    wmma_gemm_f32<1><<<dim3(SS / 64, SS / 128, BB), blk, 0, stream>>>(
        ws + OFF_QR, ws + OFF_KR, ws + OFF_SC, SS, SS, RR,
        (long long)SS * RR, (long long)SS * RR, (long long)SS * SS,
        RR, RR, SS, inv_sqrtR);

    // 4) causal top-k -> bitmask + switch-loss count
    topk_mask_kernel<<<BB * SS, 256, 0, stream>>>(ws, prev);

    // 5) q, k, v = x @ W[1024x1024]
    wmma_gemm_f32<0><<<dim3(DD / 64, (BB * SS) / 128, 1), blk, 0, stream>>>(
        x, Wq, ws + OFF_Q, BB * SS, DD, DD, 0, 0, 0, DD, DD, DD, 1.0f);
    wmma_gemm_f32<0><<<dim3(DD / 64, (BB * SS) / 128, 1), blk, 0, stream>>>(
        x, Wk, ws + OFF_K, BB * SS, DD, DD, 0, 0, 0, DD, DD, DD, 1.0f);
    wmma_gemm_f32<0><<<dim3(DD / 64, (BB * SS) / 128, 1), blk, 0, stream>>>(
        x, Wv, ws + OFF_V, BB * SS, DD, DD, 0, 0, 0, DD, DD, DD, 1.0f);

    // 6) attn_scores = q @ k^T / sqrt(D)  (overwrites scores buffer)
    wmma_gemm_f32<1><<<dim3(SS / 64, SS / 128, BB), blk, 0, stream>>>(
        ws + OFF_Q, ws + OFF_K, ws + OFF_SC, SS, SS, DD,
        (long long)SS * DD, (long long)SS * DD, (long long)SS * SS,
        DD, DD, SS, inv_sqrtD);

    // 7) masked softmax in place
    softmax_mask_kernel<<<BB * SS, 256, 0, stream>>>(ws);

    // 8) ctx = P @ v  (reuses q buffer)
    wmma_gemm_f32<0><<<dim3(DD / 64, SS / 128, BB), blk, 0, stream>>>(
        ws + OFF_SC, ws + OFF_V, ws + OFF_Q, SS, DD, SS,
        (long long)SS * SS, (long long)SS * DD, (long long)SS * DD,
        SS, DD, DD, 1.0f);

    // 9) out = ctx @ Wout
    wmma_gemm_f32<0><<<dim3(DD / 64, (BB * SS) / 128, 1), blk, 0, stream>>>(
        ws + OFF_Q, Wout, out, BB * SS, DD, DD, 0, 0, 0, DD, DD, DD, 1.0f);

    // 10) switch_loss
    finalize_loss_kernel<<<1, 1, 0, stream>>>(ws, out_loss);
}